// Net_180388626678
// MI455X (gfx1250) — compile-verified
//
#include <hip/hip_runtime.h>
#include <hip/hip_bf16.h>

// ---------------------------------------------------------------------------
// 2-layer GCN encoder + dot-product decoder for MI455X (gfx1250).
//   layer: out = D^{-1/2}(A+I)D^{-1/2} (x@W) + b
// Dense GEMMs: V_WMMA_F32_16X16X4_F32 (fp32 WMMA, K=4 per step).
// Aggregation: b128 gathers + hardware fp32 global atomics (L2-resident).
// Self-loop term folded into the elementwise bias pass (no atomics needed).
// ---------------------------------------------------------------------------

typedef __attribute__((ext_vector_type(2))) float v2f;
typedef __attribute__((ext_vector_type(8))) float v8f;

#define C1 128   // hidden width (W1 cols)
#define C2 64    // output width (W2 cols)

// ---------------------------------------------------------------- utilities
__global__ void fill_f32(float* __restrict__ p, float v, int n) {
  int i = blockIdx.x * blockDim.x + threadIdx.x;
  if (i < n) p[i] = v;
}

__global__ void zero_f32_vec4(float4* __restrict__ p, int n4) {
  int i = blockIdx.x * blockDim.x + threadIdx.x;
  if (i < n4) p[i] = make_float4(0.f, 0.f, 0.f, 0.f);
}

// deg[dst] += 1 for 4 edges per thread (deg pre-initialized to 1.0 = self loop)
__global__ void degree_accum(const int* __restrict__ dst, int E,
                             float* __restrict__ deg) {
  int i = (blockIdx.x * blockDim.x + threadIdx.x) * 4;
  if (i + 3 < E) {
    const int4 d4 = *(const int4*)&dst[i];
    __hip_atomic_fetch_add(&deg[d4.x], 1.0f, __ATOMIC_RELAXED, __HIP_MEMORY_SCOPE_AGENT);
    __hip_atomic_fetch_add(&deg[d4.y], 1.0f, __ATOMIC_RELAXED, __HIP_MEMORY_SCOPE_AGENT);
    __hip_atomic_fetch_add(&deg[d4.z], 1.0f, __ATOMIC_RELAXED, __HIP_MEMORY_SCOPE_AGENT);
    __hip_atomic_fetch_add(&deg[d4.w], 1.0f, __ATOMIC_RELAXED, __HIP_MEMORY_SCOPE_AGENT);
  } else {
    for (; i < E; ++i)
      __hip_atomic_fetch_add(&deg[dst[i]], 1.0f, __ATOMIC_RELAXED, __HIP_MEMORY_SCOPE_AGENT);
  }
}

// deg -> rsqrt(deg) in place (deg >= 1 always, self loop included)
__global__ void to_rsqrt(float* __restrict__ deg, int n) {
  int i = blockIdx.x * blockDim.x + threadIdx.x;
  if (i < n) deg[i] = rsqrtf(deg[i]);
}

// -------------------------------------------------------------- GEMM layer 1
// h1[100000,128] = x[100000,4] @ W1[4,128]; one wave per 16x16 tile, single
// f32 WMMA with K=4 (exactly the input feature width).
__global__ __launch_bounds__(32) void gemm1_wmma(const float* __restrict__ x,
                                                 const float* __restrict__ W,
                                                 float* __restrict__ h) {
  const int tm   = blockIdx.x;        // 6250 M-tiles (100000/16)
  const int tn   = blockIdx.y;        // 8 N-tiles (128/16)
  const int l    = threadIdx.x;
  const int half = l >> 4;            // lanes 16-31 carry K=2,3
  const int l15  = l & 15;
  const int kb   = half * 2;

  // A fragment: 16x4 f32, lane=row, 2 contiguous K values per lane
  const int row = tm * 16 + l15;
  const float2 av = *(const float2*)&x[row * 4 + kb];
  v2f a; a.x = av.x; a.y = av.y;

  // B fragment: 4x16 f32, lane=col, 2 K rows per lane
  const int col = tn * 16 + l15;
  v2f b; b.x = W[(kb + 0) * C1 + col];
         b.y = W[(kb + 1) * C1 + col];

  v8f c = {};
  c = __builtin_amdgcn_wmma_f32_16x16x4_f32(false, a, false, b,
                                            (short)0, c, false, false);

  // D: VGPR r -> M = r (lanes 0-15) / M = r+8 (lanes 16-31), N = lane&15
  const int mbase = tm * 16 + half * 8;
#pragma unroll
  for (int r = 0; r < 8; ++r)
    h[(mbase + r) * C1 + col] = c[r];
}

// -------------------------------------------------------------- GEMM layer 2
// h2[100000,64] = g1[100000,128] @ W2[128,64]; 32-step K-loop of f32 WMMAs.
__global__ __launch_bounds__(32) void gemm2_wmma(const float* __restrict__ g,
                                                 const float* __restrict__ W,
                                                 float* __restrict__ h2) {
  const int tm   = blockIdx.x;        // 6250 M-tiles
  const int tn   = blockIdx.y;        // 4 N-tiles (64/16)
  const int l    = threadIdx.x;
  const int half = l >> 4;
  const int l15  = l & 15;
  const int row  = tm * 16 + l15;
  const int col  = tn * 16 + l15;

  v8f c = {};
#pragma unroll 4
  for (int k = 0; k < C1; k += 4) {
    const float2 av = *(const float2*)&g[row * C1 + k + half * 2];
    v2f a; a.x = av.x; a.y = av.y;
    v2f b; b.x = W[(k + half * 2 + 0) * C2 + col];
           b.y = W[(k + half * 2 + 1) * C2 + col];
    c = __builtin_amdgcn_wmma_f32_16x16x4_f32(false, a, false, b,
                                              (short)0, c, false, false);
  }
  const int mbase = tm * 16 + half * 8;
#pragma unroll
  for (int r = 0; r < 8; ++r)
    h2[(mbase + r) * C2 + col] = c[r];
}

// ------------------------------------------------------ gather/scale/scatter
// Graph edges only (self loops handled in bias pass). One edge handled by
// C/4 threads; each thread gathers a float4 (global_load_b128) and issues 4
// fp32 global atomics to consecutive addresses.
template <int C>
__global__ __launch_bounds__(256) void scatter_edges(
    const float* __restrict__ h, const int* __restrict__ src,
    const int* __restrict__ dst, const float* __restrict__ invs,
    float* __restrict__ out, int E) {
  constexpr int TPE = C / 4;                 // threads per edge (32 or 16)
  constexpr int per_blk = 256 / TPE;         // edges per block (8 or 16)
  const long e = (long)blockIdx.x * per_blk + threadIdx.x / TPE;
  const int  j = (threadIdx.x & (TPE - 1)) * 4;
  if (e >= E) return;

  const int s = src[e];
  const int d = dst[e];
  const float norm = invs[s] * invs[d];
  const float4 hv = *(const float4*)&h[(long)s * C + j];
  float* o = &out[(long)d * C + j];
  __hip_atomic_fetch_add(o + 0, hv.x * norm, __ATOMIC_RELAXED, __HIP_MEMORY_SCOPE_AGENT);
  __hip_atomic_fetch_add(o + 1, hv.y * norm, __ATOMIC_RELAXED, __HIP_MEMORY_SCOPE_AGENT);
  __hip_atomic_fetch_add(o + 2, hv.z * norm, __ATOMIC_RELAXED, __HIP_MEMORY_SCOPE_AGENT);
  __hip_atomic_fetch_add(o + 3, hv.w * norm, __ATOMIC_RELAXED, __HIP_MEMORY_SCOPE_AGENT);
}

// ------------------------------------- bias + self-loop (+ optional ReLU)
// out[i,c] = act( agg[i,c] + h[i,c]*invs[i]^2 + b[c] )
// float4 per thread; C is a power of two so row/col are shifts/masks.
template <int C, bool RELU>
__global__ void bias_selfloop_act(float* __restrict__ g,
                                  const float* __restrict__ h,
                                  const float* __restrict__ invs,
                                  const float* __restrict__ b, int n4) {
  int i4 = blockIdx.x * blockDim.x + threadIdx.x;
  if (i4 >= n4) return;
  const long i = (long)i4 * 4;
  const int row = (int)(i / C);
  const int col = (int)(i & (C - 1));
  const float w  = invs[row];
  const float w2 = w * w;
  const float4 gv = *(const float4*)&g[i];
  const float4 hv = *(const float4*)&h[i];
  const float4 bv = *(const float4*)&b[col];
  float4 v;
  v.x = gv.x + hv.x * w2 + bv.x;
  v.y = gv.y + hv.y * w2 + bv.y;
  v.z = gv.z + hv.z * w2 + bv.z;
  v.w = gv.w + hv.w * w2 + bv.w;
  if (RELU) {
    v.x = fmaxf(v.x, 0.f); v.y = fmaxf(v.y, 0.f);
    v.z = fmaxf(v.z, 0.f); v.w = fmaxf(v.w, 0.f);
  }
  *(float4*)&g[i] = v;
}

// ------------------------------------------------------------------ decoder
// logits[e] = dot(z[s], z[d]) over 64 floats. One wave32 per edge: each lane
// takes a float2 slice, butterfly-reduce across the wave.
__global__ __launch_bounds__(256) void decode_dot(
    const float* __restrict__ z, const int* __restrict__ s_idx,
    const int* __restrict__ d_idx, float* __restrict__ out, int Elbl) {
  const int wave = threadIdx.x >> 5;
  const int lane = threadIdx.x & 31;
  const int e = blockIdx.x * 8 + wave;
  if (e >= Elbl) return;

  const int s = s_idx[e];
  const int d = d_idx[e];
  const float2 za = *(const float2*)&z[(long)s * C2 + lane * 2];
  const float2 zb = *(const float2*)&z[(long)d * C2 + lane * 2];
  float p = za.x * zb.x + za.y * zb.y;
#pragma unroll
  for (int off = 16; off > 0; off >>= 1)
    p += __shfl_xor(p, off, 32);
  if (lane == 0) out[e] = p;
}

// ---------------------------------------------------------------------------
extern "C" void kernel_launch(void* const* d_in, const int* in_sizes, int n_in,
                              void* d_out, int out_size, void* d_ws, size_t ws_size,
                              hipStream_t stream) {
  const float* x    = (const float*)d_in[0];
  const int*   ei   = (const int*)d_in[1];
  const int*   eli  = (const int*)d_in[2];
  const float* W1   = (const float*)d_in[3];
  const float* b1   = (const float*)d_in[4];
  const float* W2   = (const float*)d_in[5];
  const float* b2   = (const float*)d_in[6];
  float* logits     = (float*)d_out;

  const int N    = in_sizes[0] / 4;   // 100000
  const int E    = in_sizes[1] / 2;   // 3200000
  const int Elbl = in_sizes[2] / 2;   // 200000
  const int* src = ei;
  const int* dst = ei + E;
  const int* ls  = eli;
  const int* ld  = eli + Elbl;

  // workspace layout (bytes): invs | region A (128N f32) | region B (128N f32)
  char* ws = (char*)d_ws;
  size_t off  = 0;
  float* invs = (float*)(ws + off); off += ((size_t)N * 4 + 255) & ~(size_t)255;
  float* regA = (float*)(ws + off); off += (size_t)N * C1 * 4;   // h1, then h2
  float* regB = (float*)(ws + off);                              // g1, then z
  float* h1 = regA;
  float* g1 = regB;
  float* h2 = regA;   // reuse: h1 dead after layer-1 bias pass
  float* z  = regB;   // reuse: g1 dead after gemm2

  const int BT = 256;

  // ---- degree -> inv_sqrt (self loop folded in as init=1.0)
  fill_f32<<<(N + BT - 1) / BT, BT, 0, stream>>>(invs, 1.0f, N);
  degree_accum<<<((E + 3) / 4 + BT - 1) / BT, BT, 0, stream>>>(dst, E, invs);
  to_rsqrt<<<(N + BT - 1) / BT, BT, 0, stream>>>(invs, N);

  // ---- layer 1: h1 = x @ W1  (WMMA f32 16x16x4, K=4 exact)
  gemm1_wmma<<<dim3(N / 16, C1 / 16), 32, 0, stream>>>(x, W1, h1);

  // ---- aggregate 128-wide into g1 (graph edges), then self-loop+bias+relu
  zero_f32_vec4<<<((N * C1 / 4) + BT - 1) / BT, BT, 0, stream>>>(
      (float4*)g1, N * C1 / 4);
  {
    const int per_blk = 256 / (C1 / 4);               // 8 edges per block
    scatter_edges<C1><<<(E + per_blk - 1) / per_blk, BT, 0, stream>>>(
        h1, src, dst, invs, g1, E);
  }
  bias_selfloop_act<C1, true><<<((N * C1 / 4) + BT - 1) / BT, BT, 0, stream>>>(
      g1, h1, invs, b1, N * C1 / 4);

  // ---- layer 2: h2 = g1 @ W2  (WMMA f32 16x16x4, 32-step K loop)
  gemm2_wmma<<<dim3(N / 16, C2 / 16), 32, 0, stream>>>(g1, W2, h2);

  // ---- aggregate 64-wide into z (graph edges), then self-loop+bias
  zero_f32_vec4<<<((N * C2 / 4) + BT - 1) / BT, BT, 0, stream>>>(
      (float4*)z, N * C2 / 4);
  {
    const int per_blk = 256 / (C2 / 4);               // 16 edges per block
    scatter_edges<C2><<<(E + per_blk - 1) / per_blk, BT, 0, stream>>>(
        h2, src, dst, invs, z, E);
  }
  bias_selfloop_act<C2, false><<<((N * C2 / 4) + BT - 1) / BT, BT, 0, stream>>>(
      z, h2, invs, b2, N * C2 / 4);

  // ---- decode: one wave per labeled edge
  decode_dot<<<(Elbl + 7) / 8, BT, 0, stream>>>(z, ls, ld, logits, Elbl);
}